// FreqLoss1D_68573447848111
// MI455X (gfx1250) — compile-verified
//
#include <hip/hip_runtime.h>
#include <math.h>

// ---------------------------------------------------------------------------
// FreqLoss1D on MI455X (gfx1250)
//
// 2D FFT (1024x1024, 32 real images) via radix-32 four-step; every 32-point
// DFT batch is a 32x32x32 complex matmul on V_WMMA_F32_16X16X4_F32.
// - DFT-32 matrix in LDS in three flavors (Fr, Fi, -Fi): complex products are
//   pure WMMA accumulation chains.
// - Inter-stage twiddles come from a per-block LDS float2 table (1 b64/factor)
//   applied in the stage-A store epilogue.
// - Pass 1 exploits the real input (stage A = 64 WMMAs) and stores only the
//   first 544 frequency rows of the transposed intermediate (Hermitian
//   symmetry of a real 2D spectrum).
// - Pass 2 runs rows 0..519 only; each element's power is binned at its own
//   radius and (for 0<j<512) at its conjugate-mirror's radius, covering all
//   1024^2 pixels exactly once. The spectrum is never materialized.
// ---------------------------------------------------------------------------

typedef float v2f __attribute__((ext_vector_type(2)));
typedef float v8f __attribute__((ext_vector_type(8)));

#define SDIM 1024
#define RADIUS 512
#define NIMG 32                    // 16 "output" images then 16 "target" images
#define IMG_ELEMS (SDIM * SDIM)
#define ROWS_PER_WG 8
#define WGS_PER_IMG (SDIM / ROWS_PER_WG)   // 128 (pass 1)
#define KROWS 544                  // stored freq rows (>= 520, multiple of 32)
#define WGS_P2 65                  // pass-2 workgroups: rows 0..519
#define INTER_ELEMS ((size_t)KROWS * SDIM)

#define LDW  33                    // padded row stride of 32x32 LDS matrices
#define WBUF 1064                  // per-wave buffer floats (>= 32*LDW, 40 mod 64)

#define WMMA_F32(A, B, C) \
  __builtin_amdgcn_wmma_f32_16x16x4_f32(false, (A), false, (B), (short)0, (C), false, false)

// Fragment loaders (CDNA5 ISA 7.12.2 layouts).
//   A 16x4 : lanes 0-15 -> M=lane, VGPR0=K0 VGPR1=K1 ; lanes 16-31 -> K2,K3
//   B 4x16 : lanes 0-15 -> N=lane, VGPR0=K0 VGPR1=K1 ; lanes 16-31 -> K2,K3
//   C 16x16: VGPR r -> row r (lanes 0-15) / row r+8 (lanes 16-31), N=lane%16
__device__ __forceinline__ v2f ldA(const float* p, int m, int k0) {
  v2f a; a.x = p[m * LDW + k0]; a.y = p[m * LDW + k0 + 1]; return a;
}
__device__ __forceinline__ v2f ldB(const float* p, int k0, int n) {
  v2f b; b.x = p[k0 * LDW + n]; b.y = p[(k0 + 1) * LDW + n]; return b;
}

// Twiddle-fused store: z[row][col] = v * TWT[row][col]  (= exp(-2pi*i*r*c/1024))
__device__ __forceinline__ void st_tw(float* zr, float* zi, const float2* twt,
                                      int row, int col, float vr, float vi) {
  const float2 t = twt[row * LDW + col];              // one ds_load_b64
  zr[row * LDW + col] = vr * t.x - vi * t.y;
  zi[row * LDW + col] = vr * t.y + vi * t.x;
}

// ------- stage A, real input (pass 1): Z = tw .* (F * Ur), 64 WMMAs --------
__device__ __forceinline__ void stageA_real_tw(
    float* __restrict__ zr, float* __restrict__ zi, const float2* __restrict__ twt,
    const float* __restrict__ Fr, const float* __restrict__ Fi,
    const float* __restrict__ ur, int lane)
{
  const int laneLo = lane & 15;
  const int hiOff  = (lane >> 4) << 1;
  const int m0 = laneLo, m1 = 16 + laneLo;
  const int n0 = laneLo, n1 = 16 + laneLo;

  v8f aR00 = {0.f,0.f,0.f,0.f,0.f,0.f,0.f,0.f};
  v8f aR01 = aR00, aR10 = aR00, aR11 = aR00;
  v8f aI00 = aR00, aI01 = aR00, aI10 = aR00, aI11 = aR00;

#pragma unroll
  for (int c = 0; c < 8; ++c) {
    const int k0 = c * 4 + hiOff;
    const v2f fr0 = ldA(Fr, m0, k0), fr1 = ldA(Fr, m1, k0);
    const v2f fi0 = ldA(Fi, m0, k0), fi1 = ldA(Fi, m1, k0);
    const v2f u0  = ldB(ur, k0, n0), u1  = ldB(ur, k0, n1);
    aR00 = WMMA_F32(fr0, u0, aR00);  aI00 = WMMA_F32(fi0, u0, aI00);
    aR01 = WMMA_F32(fr0, u1, aR01);  aI01 = WMMA_F32(fi0, u1, aI01);
    aR10 = WMMA_F32(fr1, u0, aR10);  aI10 = WMMA_F32(fi1, u0, aI10);
    aR11 = WMMA_F32(fr1, u1, aR11);  aI11 = WMMA_F32(fi1, u1, aI11);
  }

  const int mB = (lane >> 4) << 3;
#pragma unroll
  for (int r = 0; r < 8; ++r) {
    st_tw(zr, zi, twt, mB + r,      n0, aR00[r], aI00[r]);
    st_tw(zr, zi, twt, mB + r,      n1, aR01[r], aI01[r]);
    st_tw(zr, zi, twt, mB + r + 16, n0, aR10[r], aI10[r]);
    st_tw(zr, zi, twt, mB + r + 16, n1, aR11[r], aI11[r]);
  }
}

// ------ stage A, complex input (pass 2): Z = tw .* (F * U), 128 WMMAs ------
__device__ __forceinline__ void stageA_cplx_tw(
    float* __restrict__ zr, float* __restrict__ zi, const float2* __restrict__ twt,
    const float* __restrict__ Fr, const float* __restrict__ Fi,
    const float* __restrict__ FiN,
    const float* __restrict__ br, const float* __restrict__ bi, int lane)
{
  const int laneLo = lane & 15;
  const int hiOff  = (lane >> 4) << 1;
  const int m0 = laneLo, m1 = 16 + laneLo;
  const int n0 = laneLo, n1 = 16 + laneLo;

  v8f aR00 = {0.f,0.f,0.f,0.f,0.f,0.f,0.f,0.f};
  v8f aR01 = aR00, aR10 = aR00, aR11 = aR00;
  v8f aI00 = aR00, aI01 = aR00, aI10 = aR00, aI11 = aR00;

#pragma unroll
  for (int c = 0; c < 8; ++c) {
    const int k0 = c * 4 + hiOff;
    const v2f fr0 = ldA(Fr,  m0, k0), fr1 = ldA(Fr,  m1, k0);
    const v2f fi0 = ldA(Fi,  m0, k0), fi1 = ldA(Fi,  m1, k0);
    const v2f fn0 = ldA(FiN, m0, k0), fn1 = ldA(FiN, m1, k0);
    const v2f ur0 = ldB(br, k0, n0), ur1 = ldB(br, k0, n1);
    const v2f ui0 = ldB(bi, k0, n0), ui1 = ldB(bi, k0, n1);
    // Re: Fr*Ur + (-Fi)*Ui ; Im: Fr*Ui + Fi*Ur
    aR00 = WMMA_F32(fr0, ur0, aR00);  aR00 = WMMA_F32(fn0, ui0, aR00);
    aI00 = WMMA_F32(fr0, ui0, aI00);  aI00 = WMMA_F32(fi0, ur0, aI00);
    aR01 = WMMA_F32(fr0, ur1, aR01);  aR01 = WMMA_F32(fn0, ui1, aR01);
    aI01 = WMMA_F32(fr0, ui1, aI01);  aI01 = WMMA_F32(fi0, ur1, aI01);
    aR10 = WMMA_F32(fr1, ur0, aR10);  aR10 = WMMA_F32(fn1, ui0, aR10);
    aI10 = WMMA_F32(fr1, ui0, aI10);  aI10 = WMMA_F32(fi1, ur0, aI10);
    aR11 = WMMA_F32(fr1, ur1, aR11);  aR11 = WMMA_F32(fn1, ui1, aR11);
    aI11 = WMMA_F32(fr1, ui1, aI11);  aI11 = WMMA_F32(fi1, ur1, aI11);
  }

  const int mB = (lane >> 4) << 3;
#pragma unroll
  for (int r = 0; r < 8; ++r) {
    st_tw(zr, zi, twt, mB + r,      n0, aR00[r], aI00[r]);
    st_tw(zr, zi, twt, mB + r,      n1, aR01[r], aI01[r]);
    st_tw(zr, zi, twt, mB + r + 16, n0, aR10[r], aI10[r]);
    st_tw(zr, zi, twt, mB + r + 16, n1, aR11[r], aI11[r]);
  }
}

// Stage-B WMMA body (shared): W = Z * F, accumulators left in registers.
#define STAGEB_BODY(ar, ai, Fr, Fi, FiN)                                      \
  const int laneLo = lane & 15;                                               \
  const int hiOff  = (lane >> 4) << 1;                                        \
  const int m0 = laneLo, m1 = 16 + laneLo;                                    \
  const int n0 = laneLo, n1 = 16 + laneLo;                                    \
  v8f aR00 = {0.f,0.f,0.f,0.f,0.f,0.f,0.f,0.f};                               \
  v8f aR01 = aR00, aR10 = aR00, aR11 = aR00;                                  \
  v8f aI00 = aR00, aI01 = aR00, aI10 = aR00, aI11 = aR00;                     \
  _Pragma("unroll")                                                           \
  for (int c = 0; c < 8; ++c) {                                               \
    const int k0 = c * 4 + hiOff;                                             \
    const v2f zr0 = ldA(ar, m0, k0), zr1 = ldA(ar, m1, k0);                   \
    const v2f zi0 = ldA(ai, m0, k0), zi1 = ldA(ai, m1, k0);                   \
    const v2f fr0 = ldB(Fr,  k0, n0), fr1 = ldB(Fr,  k0, n1);                 \
    const v2f fi0 = ldB(Fi,  k0, n0), fi1 = ldB(Fi,  k0, n1);                 \
    const v2f fn0 = ldB(FiN, k0, n0), fn1 = ldB(FiN, k0, n1);                 \
    aR00 = WMMA_F32(zr0, fr0, aR00);  aR00 = WMMA_F32(zi0, fn0, aR00);        \
    aI00 = WMMA_F32(zr0, fi0, aI00);  aI00 = WMMA_F32(zi0, fr0, aI00);        \
    aR01 = WMMA_F32(zr0, fr1, aR01);  aR01 = WMMA_F32(zi0, fn1, aR01);        \
    aI01 = WMMA_F32(zr0, fi1, aI01);  aI01 = WMMA_F32(zi0, fr1, aI01);        \
    aR10 = WMMA_F32(zr1, fr0, aR10);  aR10 = WMMA_F32(zi1, fn0, aR10);        \
    aI10 = WMMA_F32(zr1, fi0, aI10);  aI10 = WMMA_F32(zi1, fr0, aI10);        \
    aR11 = WMMA_F32(zr1, fr1, aR11);  aR11 = WMMA_F32(zi1, fn1, aR11);        \
    aI11 = WMMA_F32(zr1, fi1, aI11);  aI11 = WMMA_F32(zi1, fr1, aI11);        \
  }                                                                           \
  const int mB = (lane >> 4) << 3;

// ---- stage B + store (pass 1): slot [k2][k1] <- X[k2 + 32*k1] -------------
__device__ __forceinline__ void stageB_store(
    float* __restrict__ cr, float* __restrict__ ci,
    const float* __restrict__ ar, const float* __restrict__ ai,
    const float* __restrict__ Fr, const float* __restrict__ Fi,
    const float* __restrict__ FiN, int lane)
{
  STAGEB_BODY(ar, ai, Fr, Fi, FiN)
#pragma unroll
  for (int r = 0; r < 8; ++r) {
    cr[(mB + r)      * LDW + n0] = aR00[r];  ci[(mB + r)      * LDW + n0] = aI00[r];
    cr[(mB + r)      * LDW + n1] = aR01[r];  ci[(mB + r)      * LDW + n1] = aI01[r];
    cr[(mB + r + 16) * LDW + n0] = aR10[r];  ci[(mB + r + 16) * LDW + n0] = aI10[r];
    cr[(mB + r + 16) * LDW + n1] = aR11[r];  ci[(mB + r + 16) * LDW + n1] = aI11[r];
  }
}

// ---- stage B + radial binning (pass 2) ------------------------------------
// Bins each element's power at its own radius (j <= 512) and, for 0<j<512,
// also at the radius of its Hermitian mirror (1024-j, (1024-m)%1024), whose
// power is identical for a real input. Covers all 1024^2 pixels exactly once.
__device__ __forceinline__ void bin_el(float* lbins, float cj2, float cjm2,
                                       bool doMain, bool doMirror,
                                       int row, int col, float vr, float vi) {
  const int   m = row + (col << 5);                   // freq = k2 + 32*k1
  const float p = vr * vr + vi * vi;
  if (doMain) {
    const float cm = (m < 512) ? (m + 0.5f) : (m - 1023.5f);
    const int ib = (int)sqrtf(cj2 + cm * cm);
    if (ib < RADIUS) atomicAdd(&lbins[ib], p);        // ds_add_f32
  }
  if (doMirror) {
    const int   m2  = (1024 - m) & 1023;
    const float cm2 = (m2 < 512) ? (m2 + 0.5f) : (m2 - 1023.5f);
    const int ib2 = (int)sqrtf(cjm2 + cm2 * cm2);
    if (ib2 < RADIUS) atomicAdd(&lbins[ib2], p);
  }
}

__device__ __forceinline__ void stageB_bin(
    const float* __restrict__ ar, const float* __restrict__ ai,
    const float* __restrict__ Fr, const float* __restrict__ Fi,
    const float* __restrict__ FiN, int lane,
    float cj2, float cjm2, bool doMain, bool doMirror, float* lbins)
{
  STAGEB_BODY(ar, ai, Fr, Fi, FiN)
#pragma unroll
  for (int r = 0; r < 8; ++r) {
    bin_el(lbins, cj2, cjm2, doMain, doMirror, mB + r,      n0, aR00[r], aI00[r]);
    bin_el(lbins, cj2, cjm2, doMain, doMirror, mB + r,      n1, aR01[r], aI01[r]);
    bin_el(lbins, cj2, cjm2, doMain, doMirror, mB + r + 16, n0, aR10[r], aI10[r]);
    bin_el(lbins, cj2, cjm2, doMain, doMirror, mB + r + 16, n1, aR11[r], aI11[r]);
  }
}

// Per-block constant tables: DFT-32 matrix (3 flavors) + 32x32 twiddle table.
__device__ __forceinline__ void build_tables(float* Fr, float* Fi, float* FiN,
                                             float2* twt, int tid)
{
  for (int q = tid; q < 1024; q += 256) {
    const int i = q >> 5, j = q & 31;
    const int pos = i * LDW + j;
    float sn, cs;
    // F32[k][n] = exp(-2*pi*i*k*n/32), symmetric
    __sincosf(-0.19634954084936207f * (float)((i * j) & 31), &sn, &cs);
    Fr[pos]  = cs;
    Fi[pos]  = sn;
    FiN[pos] = -sn;
    // twiddle: exp(-2*pi*i*i*j/1024)
    __sincosf(-6.1359231515425649e-3f * (float)(i * j), &sn, &cs);
    twt[pos].x = cs;
    twt[pos].y = sn;
  }
}

// --------------------------- pass 1: row FFTs ------------------------------
// One wave per image row (real input); freq rows k < KROWS stored TRANSPOSED
// into the intermediate (inter[k][row]); the rest is Hermitian-redundant.
__global__ __launch_bounds__(256) void pass_rows(
    const float* __restrict__ outImgs, const float* __restrict__ tgtImgs,
    float2* __restrict__ inter, int baseImg)
{
  __shared__ float  Fr[32 * LDW], Fi[32 * LDW], FiN[32 * LDW];
  __shared__ float2 TWT[32 * LDW];
  __shared__ float  XR[8][WBUF], XI[8][WBUF];
  __shared__ float  ZR[8][WBUF], ZI[8][WBUF];

  const int tid  = threadIdx.x;
  const int lane = tid & 31;
  const int wave = tid >> 5;
  const int img  = baseImg + blockIdx.y;
  const int rowbase = blockIdx.x * ROWS_PER_WG;
  const int row  = rowbase + wave;

  build_tables(Fr, Fi, FiN, TWT, tid);

  const float* src = (img < 16) ? (outImgs + (size_t)img * IMG_ELEMS)
                                : (tgtImgs + (size_t)(img - 16) * IMG_ELEMS);
  const float* rowp = src + (size_t)row * SDIM;
  for (int q = lane; q < 1024; q += 32)
    XR[wave][(q >> 5) * LDW + (q & 31)] = rowp[q];

  __syncthreads();                                   // tables + inputs ready

  stageA_real_tw(ZR[wave], ZI[wave], TWT, Fr, Fi, XR[wave], lane);
  stageB_store(XR[wave], XI[wave], ZR[wave], ZI[wave], Fr, Fi, FiN, lane);

  __syncthreads();                                   // all 8 rows finished

  // Cooperative transposed store of freq rows k < KROWS:
  // 8 consecutive image rows -> 64B contiguous chunks.
  float2* dst = inter + (size_t)blockIdx.y * INTER_ELEMS;
  for (int kb = 0; kb < KROWS; kb += 32) {
    const int k = kb + (tid >> 3);
    const int r = tid & 7;
    const int q = (k & 31) * LDW + (k >> 5);         // digit-reversed slot
    float2 v;
    v.x = XR[r][q];
    v.y = XI[r][q];
    dst[(size_t)k * SDIM + rowbase + r] = v;
  }
}

// ------------------- pass 2: column FFTs + radial binning ------------------
__global__ __launch_bounds__(256) void pass_cols(
    const float2* __restrict__ inter, float* __restrict__ bins, int baseImg)
{
  __shared__ float  Fr[32 * LDW], Fi[32 * LDW], FiN[32 * LDW];
  __shared__ float2 TWT[32 * LDW];
  __shared__ float  XR[8][WBUF], XI[8][WBUF];
  __shared__ float  ZR[8][WBUF], ZI[8][WBUF];
  __shared__ float  lbins[RADIUS];

  const int tid  = threadIdx.x;
  const int lane = tid & 31;
  const int wave = tid >> 5;
  const int img  = baseImg + blockIdx.y;
  const int j    = blockIdx.x * ROWS_PER_WG + wave;  // freq of first FFT dim

  build_tables(Fr, Fi, FiN, TWT, tid);
  for (int q = tid; q < RADIUS; q += 256) lbins[q] = 0.0f;

  const float2* rp = inter + (size_t)blockIdx.y * INTER_ELEMS + (size_t)j * SDIM;
  for (int q = lane; q < 1024; q += 32) {
    const float2 v = rp[q];
    const int pos = (q >> 5) * LDW + (q & 31);
    XR[wave][pos] = v.x;
    XI[wave][pos] = v.y;
  }

  __syncthreads();                                   // tables + lbins ready

  // fftshifted coordinate of a frequency index f: (f<512) ? f+0.5 : f-1023.5
  const int   jm  = (1024 - j) & 1023;               // Hermitian mirror row
  const float cj  = (j  < 512) ? (j  + 0.5f) : (j  - 1023.5f);
  const float cjm = (jm < 512) ? (jm + 0.5f) : (jm - 1023.5f);
  const bool  doMain   = (j <= 512);                 // rows 513..519 are padding
  const bool  doMirror = (j > 0) && (j < 512);       // rows 0/512 are self-mirror

  stageA_cplx_tw(ZR[wave], ZI[wave], TWT, Fr, Fi, FiN, XR[wave], XI[wave], lane);
  stageB_bin(ZR[wave], ZI[wave], Fr, Fi, FiN, lane,
             cj * cj, cjm * cjm, doMain, doMirror, lbins);

  __syncthreads();                                   // all binning atomics done
  for (int b = tid; b < RADIUS; b += 256)
    atomicAdd(&bins[(size_t)img * RADIUS + b], lbins[b]);
}

// --------------------------- small helpers ---------------------------------
__global__ void zero_bins(float* bins)
{
  const int i = blockIdx.x * 256 + threadIdx.x;
  if (i < NIMG * RADIUS) bins[i] = 0.0f;
}

__global__ __launch_bounds__(256) void final_reduce(
    const float* __restrict__ bins, float* __restrict__ out)
{
  __shared__ float red[256];
  float s = 0.0f;
  for (int i = threadIdx.x; i < 16 * RADIUS; i += 256)
    s += fabsf(bins[i] - bins[16 * RADIUS + i]);
  red[threadIdx.x] = s;
  __syncthreads();
  for (int off = 128; off > 0; off >>= 1) {
    if (threadIdx.x < off) red[threadIdx.x] += red[threadIdx.x + off];
    __syncthreads();
  }
  if (threadIdx.x == 0)
    out[0] = red[0] * (1.0f / (float)IMG_ELEMS) * (1.0f / (16.0f * RADIUS));
}

// ------------------------------- launcher ----------------------------------
extern "C" void kernel_launch(void* const* d_in, const int* in_sizes, int n_in,
                              void* d_out, int out_size, void* d_ws, size_t ws_size,
                              hipStream_t stream) {
  (void)in_sizes; (void)n_in; (void)out_size;
  const float* outImgs = (const float*)d_in[0];   // [16,1,1024,1024] f32
  const float* tgtImgs = (const float*)d_in[1];   // [16,1,1024,1024] f32

  float* bins = (float*)d_ws;                      // 32*512 f32 = 64 KB
  const size_t binsBytes  = (size_t)NIMG * RADIUS * sizeof(float);
  float2* inter = (float2*)((char*)d_ws + binsBytes);
  const size_t perImg = INTER_ELEMS * sizeof(float2);   // 4.25 MB

  // Batch as many images as the workspace allows (deterministic for fixed ws).
  size_t avail = (ws_size > binsBytes) ? (ws_size - binsBytes) : 0;
  int batch = (int)(avail / perImg);
  if (batch < 1) batch = 1;
  if (batch > NIMG) batch = NIMG;

  zero_bins<<<(NIMG * RADIUS + 255) / 256, 256, 0, stream>>>(bins);

  for (int base = 0; base < NIMG; base += batch) {
    const int b = (NIMG - base < batch) ? (NIMG - base) : batch;
    pass_rows<<<dim3(WGS_PER_IMG, b), 256, 0, stream>>>(outImgs, tgtImgs, inter, base);
    pass_cols<<<dim3(WGS_P2,      b), 256, 0, stream>>>(inter, bins, base);
  }

  final_reduce<<<1, 256, 0, stream>>>(bins, (float*)d_out);
}